// Decoder_25048249270382
// MI455X (gfx1250) — compile-verified
//
#include <hip/hip_runtime.h>

typedef __attribute__((ext_vector_type(2))) float v2f;
typedef __attribute__((ext_vector_type(8))) float v8f;

// ---------------------------------------------------------------------------
// Degree / normalization kernels
// ---------------------------------------------------------------------------
__global__ void k_init_one(float* __restrict__ deg, int n) {
    int i = blockIdx.x * blockDim.x + threadIdx.x;
    if (i < n) deg[i] = 1.0f;  // self-loop contributes 1 to every node's degree
}

__global__ void k_deg_accum(const int* __restrict__ dst, float* __restrict__ deg, int e) {
    int i = blockIdx.x * blockDim.x + threadIdx.x;
    if (i < e) atomicAdd(&deg[dst[i]], 1.0f);
}

__global__ void k_rsqrt_inplace(float* __restrict__ deg, int n) {
    int i = blockIdx.x * blockDim.x + threadIdx.x;
    if (i < n) deg[i] = rsqrtf(deg[i]);  // deg >= 1 always (self-loops)
}

// ---------------------------------------------------------------------------
// WMMA fp32 GEMM: C[M x N] = A[M x K] @ B[K x N] (+ bias, optional relu)
// One 16x16 output tile per wave32; K unrolled in steps of 4 via
// V_WMMA_F32_16X16X4_F32. Layouts per CDNA5 ISA 7.12.2:
//   A (16x4 f32, 2 VGPR): lane m=lane&15; V0 holds K=k0+2*half, V1 holds K+1
//   B (4x16 f32, 2 VGPR): lane n=lane&15; V0 holds K=k0+2*half, V1 holds K+1
//   C/D (16x16 f32, 8 VGPR): VGPR v -> row v + 8*half, col = lane&15
// ---------------------------------------------------------------------------
template <int K, int N>
__global__ void k_gemm_wmma(const float* __restrict__ A, const float* __restrict__ B,
                            const float* __restrict__ bias, float* __restrict__ C,
                            int M, int relu) {
    const int lane = threadIdx.x & 31;
    const int wave = threadIdx.x >> 5;
    const int half = lane >> 4;   // 0: lanes 0-15, 1: lanes 16-31
    const int lm   = lane & 15;

    const int tilesN = N >> 4;
    const int tile = blockIdx.x * (blockDim.x >> 5) + wave;
    const int totalTiles = (M >> 4) * tilesN;
    if (tile >= totalTiles) return;  // wave-uniform exit; EXEC stays all-1s per wave

    const int row0 = (tile / tilesN) << 4;
    const int n0   = (tile % tilesN) << 4;

    v8f acc = {};
#pragma unroll
    for (int k0 = 0; k0 < K; k0 += 4) {
        const int ka = k0 + 2 * half;
        v2f a, b;
        a.x = A[(size_t)(row0 + lm) * K + ka];
        a.y = A[(size_t)(row0 + lm) * K + ka + 1];
        b.x = B[(size_t)ka * N + n0 + lm];
        b.y = B[(size_t)(ka + 1) * N + n0 + lm];
        acc = __builtin_amdgcn_wmma_f32_16x16x4_f32(
            false, a, false, b, (short)0, acc, false, false);
    }

    const float bv = bias ? bias[n0 + lm] : 0.0f;
#pragma unroll
    for (int v = 0; v < 8; ++v) {
        const int row = row0 + v + 8 * half;
        float val = acc[v] + bv;
        if (relu) val = fmaxf(val, 0.0f);
        C[(size_t)row * N + n0 + lm] = val;
    }
}

// ---------------------------------------------------------------------------
// Aggregation: out[i,f] = h[i,f] * dinv[i]^2  (self-loop term, non-atomic init)
// ---------------------------------------------------------------------------
__global__ void k_self_init(const float* __restrict__ h, const float* __restrict__ dinv,
                            float* __restrict__ out, long long total, int f) {
    long long idx = (long long)blockIdx.x * blockDim.x + threadIdx.x;
    if (idx >= total) return;
    int node = (int)(idx / f);
    float d = dinv[node];
    out[idx] = h[idx] * d * d;
}

// out[dst,f] += h[src,f] * dinv[src] * dinv[dst]   (one thread per edge-feature)
__global__ void k_edge_agg(const float* __restrict__ h, const int* __restrict__ src,
                           const int* __restrict__ dst, const float* __restrict__ dinv,
                           float* __restrict__ out, long long total, int f) {
    long long idx = (long long)blockIdx.x * blockDim.x + threadIdx.x;
    if (idx >= total) return;
    int e  = (int)(idx / f);
    int ff = (int)(idx % f);
    int s = src[e], d = dst[e];
    float val = h[(size_t)s * f + ff] * dinv[s] * dinv[d];
    atomicAdd(&out[(size_t)d * f + ff], val);
}

// x = (relu?) (x + b[col])
__global__ void k_bias_act(float* __restrict__ x, const float* __restrict__ b,
                           long long total, int f, int relu) {
    long long idx = (long long)blockIdx.x * blockDim.x + threadIdx.x;
    if (idx >= total) return;
    float v = x[idx] + b[(int)(idx % f)];
    if (relu) v = fmaxf(v, 0.0f);
    x[idx] = v;
}

// ---------------------------------------------------------------------------
// Host-side launch
// ---------------------------------------------------------------------------
static inline int cdiv(long long a, int b) { return (int)((a + b - 1) / b); }

extern "C" void kernel_launch(void* const* d_in, const int* in_sizes, int n_in,
                              void* d_out, int out_size, void* d_ws, size_t ws_size,
                              hipStream_t stream) {
    (void)n_in; (void)out_size; (void)ws_size;

    const float* z   = (const float*)d_in[0];
    const int*   ei  = (const int*)  d_in[1];
    const float* W1  = (const float*)d_in[2];
    const float* b1  = (const float*)d_in[3];
    const float* W2  = (const float*)d_in[4];
    const float* b2  = (const float*)d_in[5];
    const float* Wfc = (const float*)d_in[6];
    const float* bfc = (const float*)d_in[7];
    float* out = (float*)d_out;

    const int IN_C  = 64;                    // z feature dim
    const int HID   = in_sizes[3];           // 64
    const int OUT_C = in_sizes[5];           // 32
    const int NODES = in_sizes[0] / IN_C;    // 50000
    const int E     = in_sizes[1] / 2;       // 800000
    const int* srcIdx = ei;                  // edge_index[0]
    const int* dstIdx = ei + E;              // edge_index[1]

    // Workspace layout (floats), 64-element aligned regions
    float* ws = (float*)d_ws;
    size_t off = 0;
    float* dinv = ws + off; off += (size_t)NODES;         off = (off + 63) & ~(size_t)63;
    float* h1   = ws + off; off += (size_t)NODES * HID;   off = (off + 63) & ~(size_t)63;
    float* a1   = ws + off; off += (size_t)NODES * HID;   off = (off + 63) & ~(size_t)63;
    float* h2   = ws + off; off += (size_t)NODES * OUT_C; off = (off + 63) & ~(size_t)63;
    float* a2   = ws + off; off += (size_t)NODES * OUT_C;

    const int T = 256;

    // ---- normalization: deg -> dinv ----
    k_init_one<<<cdiv(NODES, T), T, 0, stream>>>(dinv, NODES);
    k_deg_accum<<<cdiv(E, T), T, 0, stream>>>(dstIdx, dinv, E);
    k_rsqrt_inplace<<<cdiv(NODES, T), T, 0, stream>>>(dinv, NODES);

    // ---- layer 1: h1 = z @ W1 ; a1 = relu(Aggregate(h1) + b1) ----
    {
        int waves = (NODES / 16) * (HID / 16);
        k_gemm_wmma<64, 64><<<cdiv(waves, 4), 128, 0, stream>>>(z, W1, nullptr, h1, NODES, 0);
        long long tn = (long long)NODES * HID;
        k_self_init<<<cdiv(tn, T), T, 0, stream>>>(h1, dinv, a1, tn, HID);
        long long te = (long long)E * HID;
        k_edge_agg<<<cdiv(te, T), T, 0, stream>>>(h1, srcIdx, dstIdx, dinv, a1, te, HID);
        k_bias_act<<<cdiv(tn, T), T, 0, stream>>>(a1, b1, tn, HID, 1);
    }

    // ---- layer 2: h2 = a1 @ W2 ; a2 = Aggregate(h2) + b2 ----
    {
        int waves = (NODES / 16) * (OUT_C / 16);
        k_gemm_wmma<64, 32><<<cdiv(waves, 4), 128, 0, stream>>>(a1, W2, nullptr, h2, NODES, 0);
        long long tn = (long long)NODES * OUT_C;
        k_self_init<<<cdiv(tn, T), T, 0, stream>>>(h2, dinv, a2, tn, OUT_C);
        long long te = (long long)E * OUT_C;
        k_edge_agg<<<cdiv(te, T), T, 0, stream>>>(h2, srcIdx, dstIdx, dinv, a2, te, OUT_C);
        k_bias_act<<<cdiv(tn, T), T, 0, stream>>>(a2, b2, tn, OUT_C, 0);
    }

    // ---- FC head: out = a2 @ Wfc + bfc  (bias fused into WMMA epilogue) ----
    {
        int waves = (NODES / 16) * (1024 / 16);
        k_gemm_wmma<32, 1024><<<cdiv(waves, 4), 128, 0, stream>>>(a2, Wfc, bfc, out, NODES, 0);
    }
}